// MaskLoss_80212809220305
// MI455X (gfx1250) — compile-verified
//
#include <hip/hip_runtime.h>
#include <hip/hip_bf16.h>
#include <math.h>

typedef __attribute__((ext_vector_type(2))) float v2f;
typedef __attribute__((ext_vector_type(8))) float v8f;

#define WMMA_F32(A, B, C) \
  __builtin_amdgcn_wmma_f32_16x16x4_f32(false, (A), false, (B), (short)0, (C), false, false)

// ---------------------------------------------------------------------------
// Kernel 1: per-filter normalization  nf[n][d] = (w[n][d]-mean)/std (std0->1)
// ---------------------------------------------------------------------------
__global__ __launch_bounds__(256) void nf_normalize_kernel(
    const float* __restrict__ w, float* __restrict__ nf, int D) {
  __shared__ float s_sum[256];
  __shared__ float s_sq[256];
  const int n = blockIdx.x;
  const float* row = w + (size_t)n * (size_t)D;
  float s = 0.f, sq = 0.f;
  for (int d = threadIdx.x; d < D; d += 256) {
    float v = row[d];
    s += v;
    sq += v * v;
  }
  s_sum[threadIdx.x] = s;
  s_sq[threadIdx.x] = sq;
  __syncthreads();
  for (int off = 128; off > 0; off >>= 1) {
    if ((int)threadIdx.x < off) {
      s_sum[threadIdx.x] += s_sum[threadIdx.x + off];
      s_sq[threadIdx.x]  += s_sq[threadIdx.x + off];
    }
    __syncthreads();
  }
  const float invDf = 1.0f / (float)D;
  const float mean = s_sum[0] * invDf;
  float var = s_sq[0] * invDf - mean * mean;   // ddof=0
  var = var < 0.f ? 0.f : var;
  const float stdv = sqrtf(var);
  const float inv = (stdv == 0.f) ? 1.f : (1.f / stdv);
  float* orow = nf + (size_t)n * (size_t)D;
  for (int d = threadIdx.x; d < D; d += 256) {
    orow[d] = (row[d] - mean) * inv;
  }
}

// ---------------------------------------------------------------------------
// Kernel 2: zero squared-sum accumulator; reduce mask sum S.
// accum[0] = 0 (squared sum), accum[1] = S
// ---------------------------------------------------------------------------
__global__ __launch_bounds__(256) void prep_kernel(
    const float* __restrict__ mask, int N, float* __restrict__ accum) {
  __shared__ float s_sum[256];
  float t = 0.f;
  for (int i = threadIdx.x; i < N; i += 256) t += mask[i];
  s_sum[threadIdx.x] = t;
  __syncthreads();
  for (int off = 128; off > 0; off >>= 1) {
    if ((int)threadIdx.x < off) s_sum[threadIdx.x] += s_sum[threadIdx.x + off];
    __syncthreads();
  }
  if (threadIdx.x == 0) {
    accum[0] = 0.f;
    accum[1] = s_sum[0];
  }
}

// ---------------------------------------------------------------------------
// Epilogue helper: contribution of one 16x16 f32 C/D tile.
// C/D layout (ISA 7.12.2): VGPR e, lanes 0-15 -> M=e, lanes 16-31 -> M=e+8,
// N = lane & 15.
// ---------------------------------------------------------------------------
__device__ __forceinline__ float tile_contrib(v8f acc, int rowTile, int colTile,
                                              int rowBase, int colL,
                                              const float* __restrict__ mask,
                                              float invD) {
  float s = 0.f;
#pragma unroll
  for (int e = 0; e < 8; ++e) {
    int rg = rowTile + rowBase + e;
    int cg = colTile + colL;
    if (cg > rg) {  // strict upper triangle (triu k=1)
      float v = acc[e] * invD * mask[rg] * mask[cg];
      s += v * v;
    }
  }
  return s;
}

// ---------------------------------------------------------------------------
// Kernel 3: upper-triangular blocked Gram via V_WMMA_F32_16X16X4_F32.
// Block = 64x64 output, 2 waves; each wave computes 32 rows x 64 cols as a
// 2x4 arrangement of 16x16 WMMA tiles (8 accumulators, 8 WMMA : 6 loads per
// K-step of 4 => ~10.7 FLOP per L2 byte). Explicit double-buffered operand
// prefetch keeps a full K-step of loads in flight behind the WMMAs.
//
// f32 operand layouts (ISA 7.12.2): A 16x4: lane&15 = M, lanes>=16 hold
// K=2,3 (VGPR0=K0/K2, VGPR1=K1/K3); B 4x16 mirrored, so A and B loads share
// one float2 gather pattern with different base rows.
// ---------------------------------------------------------------------------
__global__ __launch_bounds__(64) void gram_kernel(
    const float* __restrict__ nf, const float* __restrict__ mask,
    float* __restrict__ accum, int D) {
  const int bc = blockIdx.x;   // column block (64 wide)
  const int br = blockIdx.y;   // row block (64 tall)
  if (br > bc) return;         // only upper-triangular blocks (block-uniform)

  const int tid  = threadIdx.x;
  const int wave = tid >> 5;          // 0..1
  const int lane = tid & 31;
  const int i0 = br * 64 + wave * 32; // wave's row base (2 row tiles)
  const int j0 = bc * 64;             // wave's col base (4 col tiles)

  const int r  = lane & 15;           // M (or N) index within tile
  const int kc = (lane >> 4) << 1;    // K sub-offset: 0 or 2

  const float* pa0 = nf + (size_t)(i0 + r)      * (size_t)D + kc;
  const float* pa1 = nf + (size_t)(i0 + 16 + r) * (size_t)D + kc;
  const float* pb0 = nf + (size_t)(j0 + r)      * (size_t)D + kc;
  const float* pb1 = nf + (size_t)(j0 + 16 + r) * (size_t)D + kc;
  const float* pb2 = nf + (size_t)(j0 + 32 + r) * (size_t)D + kc;
  const float* pb3 = nf + (size_t)(j0 + 48 + r) * (size_t)D + kc;

  v8f acc00 = {}, acc01 = {}, acc02 = {}, acc03 = {};
  v8f acc10 = {}, acc11 = {}, acc12 = {}, acc13 = {};

  // Prime the double buffer.
  v2f a0 = *(const v2f*)(pa0);
  v2f a1 = *(const v2f*)(pa1);
  v2f b0 = *(const v2f*)(pb0);
  v2f b1 = *(const v2f*)(pb1);
  v2f b2 = *(const v2f*)(pb2);
  v2f b3 = *(const v2f*)(pb3);

  for (int k = 0; k < D - 4; k += 4) {  // D = 9216, divisible by 4
    // Prefetch next K-step while current WMMAs execute.
    v2f a0n = *(const v2f*)(pa0 + k + 4);
    v2f a1n = *(const v2f*)(pa1 + k + 4);
    v2f b0n = *(const v2f*)(pb0 + k + 4);
    v2f b1n = *(const v2f*)(pb1 + k + 4);
    v2f b2n = *(const v2f*)(pb2 + k + 4);
    v2f b3n = *(const v2f*)(pb3 + k + 4);

    acc00 = WMMA_F32(a0, b0, acc00);
    acc01 = WMMA_F32(a0, b1, acc01);
    acc02 = WMMA_F32(a0, b2, acc02);
    acc03 = WMMA_F32(a0, b3, acc03);
    acc10 = WMMA_F32(a1, b0, acc10);
    acc11 = WMMA_F32(a1, b1, acc11);
    acc12 = WMMA_F32(a1, b2, acc12);
    acc13 = WMMA_F32(a1, b3, acc13);

    a0 = a0n; a1 = a1n;
    b0 = b0n; b1 = b1n; b2 = b2n; b3 = b3n;
  }
  // Peeled final K-step.
  acc00 = WMMA_F32(a0, b0, acc00);
  acc01 = WMMA_F32(a0, b1, acc01);
  acc02 = WMMA_F32(a0, b2, acc02);
  acc03 = WMMA_F32(a0, b3, acc03);
  acc10 = WMMA_F32(a1, b0, acc10);
  acc11 = WMMA_F32(a1, b1, acc11);
  acc12 = WMMA_F32(a1, b2, acc12);
  acc13 = WMMA_F32(a1, b3, acc13);

  // Epilogue: corr = G/D, mask, strict-upper, square, reduce.
  const float invD    = 1.0f / (float)D;
  const int   colL    = lane & 15;
  const int   rowBase = (lane >> 4) * 8;

  float part = 0.f;
  part += tile_contrib(acc00, i0,      j0,      rowBase, colL, mask, invD);
  part += tile_contrib(acc01, i0,      j0 + 16, rowBase, colL, mask, invD);
  part += tile_contrib(acc02, i0,      j0 + 32, rowBase, colL, mask, invD);
  part += tile_contrib(acc03, i0,      j0 + 48, rowBase, colL, mask, invD);
  part += tile_contrib(acc10, i0 + 16, j0,      rowBase, colL, mask, invD);
  part += tile_contrib(acc11, i0 + 16, j0 + 16, rowBase, colL, mask, invD);
  part += tile_contrib(acc12, i0 + 16, j0 + 32, rowBase, colL, mask, invD);
  part += tile_contrib(acc13, i0 + 16, j0 + 48, rowBase, colL, mask, invD);

  __shared__ float sred[64];
  sred[tid] = part;
  __syncthreads();
  for (int off = 32; off > 0; off >>= 1) {
    if (tid < off) sred[tid] += sred[tid + off];
    __syncthreads();
  }
  if (tid == 0) atomicAdd(&accum[0], sred[0]);
}

// ---------------------------------------------------------------------------
// Kernel 4: loss = squared_sum / (S*(S-1)/2), or 0 if no active pairs.
// ---------------------------------------------------------------------------
__global__ void finalize_kernel(const float* __restrict__ accum,
                                float* __restrict__ out) {
  float S = accum[1];
  float na = 0.5f * S * (S - 1.0f);   // sum of strict-upper mask products
  out[0] = (na > 0.f) ? (accum[0] / na) : 0.f;
}

// ---------------------------------------------------------------------------
extern "C" void kernel_launch(void* const* d_in, const int* in_sizes, int n_in,
                              void* d_out, int out_size, void* d_ws, size_t ws_size,
                              hipStream_t stream) {
  const float* w    = (const float*)d_in[0];  // [N, C, KH, KW] fp32
  const float* mask = (const float*)d_in[1];  // [N, 1, 1, 1]   fp32
  float* out = (float*)d_out;

  const int total = in_sizes[0];
  const int N = in_sizes[1];
  const int D = total / N;                    // 9216

  // ws layout: nf [N*D floats], then accum [2 floats]
  float* nf    = (float*)d_ws;
  float* accum = (float*)((char*)d_ws + (size_t)N * (size_t)D * sizeof(float));

  nf_normalize_kernel<<<N, 256, 0, stream>>>(w, nf, D);
  prep_kernel<<<1, 256, 0, stream>>>(mask, N, accum);

  const int nb = N / 64;                      // 32 -> 528 live blocks
  dim3 grid(nb, nb);
  gram_kernel<<<grid, 64, 0, stream>>>(nf, mask, accum, D);

  finalize_kernel<<<1, 1, 0, stream>>>(accum, out);
}